// _AnchorTargetLayer_34093450395666
// MI455X (gfx1250) — compile-verified
//
#include <hip/hip_runtime.h>
#include <stdint.h>

// ---------------- problem constants (fixed by reference setup) ----------------
constexpr int   B_      = 2;
constexpr int   H_      = 48;
constexpr int   W_      = 48;
constexpr int   D_      = 24;
constexpr int   K_      = 20;
constexpr int   A_      = 9;
constexpr int   TOTAL   = H_ * W_ * D_ * A_;      // 497664
constexpr int   GTN     = K_ * 7;                 // 140 floats of gt boxes per batch
constexpr float NEG_OV  = 0.3f;
constexpr float POS_OV  = 0.7f;
constexpr int   RPN_BATCH = 256;
constexpr int   NUM_FG  = 128;
constexpr int   FG_CAP  = 65536;                  // compacted fg list capacity / batch
constexpr int   NBUCK   = 1024;                   // bg histogram buckets
static_assert(TOTAL % 256 == 0, "grid sizing");

#ifndef __has_builtin
#define __has_builtin(x) 0
#endif

// ---------------- CDNA5 TDM: tensor_load_to_lds (6-arg clang-23 form) ---------
#if __has_builtin(__builtin_amdgcn_tensor_load_to_lds)
#define ATL_HAVE_TDM 1
typedef unsigned int atl_u32x4 __attribute__((ext_vector_type(4)));
typedef int          atl_i32x8 __attribute__((ext_vector_type(8)));
typedef int          atl_i32x4 __attribute__((ext_vector_type(4)));

// 1-D f32 tile load: global -> LDS, per CDNA5 ISA D# layout (CDNA5 ISA)
__device__ __forceinline__ void atl_tdm_load_f32(const void* gptr,
                                                 unsigned lds_byte_off,
                                                 unsigned nelem) {
  unsigned long long ga = (unsigned long long)(uintptr_t)gptr;
  atl_u32x4 g0;
  g0.x = 1u;                                                  // count=1 (valid), no gather
  g0.y = lds_byte_off;                                        // lds_addr
  g0.z = (unsigned)(ga & 0xffffffffull);                      // global_addr[31:0]
  g0.w = (unsigned)((ga >> 32) & 0x01ffffffull) | (2u << 30); // addr[56:32] | type=2
  atl_i32x8 g1;
  g1[0] = (int)(2u << 16);                          // data_size=2 (4B), mask/flags=0
  g1[1] = (int)((nelem & 0xffffu) << 16);           // tensor_dim0[15:0] @bits63:48
  g1[2] = (int)(((nelem >> 16) & 0xffffu) | (1u << 16)); // td0[31:16] | tensor_dim1=1
  g1[3] = (int)((nelem & 0xffffu) << 16);           // td1[31:16]=0 | tile_dim0=nelem
  g1[4] = 1;                                        // tile_dim1=1, tile_dim2=0
  g1[5] = (int)nelem;                               // tensor_dim0_stride[31:0]
  g1[6] = 0;
  g1[7] = 0;
  atl_i32x4 z4 = {0, 0, 0, 0};                      // D# group 2 (NULL, <=2D tensor)
  atl_i32x8 z8 = {0, 0, 0, 0, 0, 0, 0, 0};          // trailing group (NULL)
  __builtin_amdgcn_tensor_load_to_lds(g0, g1, z4, z4, z8, 0);
}
#else
#define ATL_HAVE_TDM 0
#endif

// Stage N f32 from global into LDS offset 0 (sh struct's first member).
template <int N>
__device__ __forceinline__ void load_gt_to_lds(const float* gsrc, float* sdst,
                                               int tid, int nthreads) {
#if ATL_HAVE_TDM
  (void)sdst; (void)nthreads;
  if (tid == 0) atl_tdm_load_f32(gsrc, 0u, (unsigned)N);
  __builtin_amdgcn_s_wait_tensorcnt(0);
#else
  for (int i = tid; i < N; i += nthreads) sdst[i] = gsrc[i];
#endif
}

// ---------------- ordered-uint float max encoding -----------------------------
__device__ __forceinline__ unsigned fkey(float f) {
  unsigned u = __float_as_uint(f);
  return (u & 0x80000000u) ? ~u : (u | 0x80000000u);
}
__device__ __forceinline__ float funkey(unsigned k) {
  unsigned u = (k & 0x80000000u) ? (k & 0x7fffffffu) : ~k;
  return __uint_as_float(u);
}

// ---------------- geometry helpers --------------------------------------------
struct AnchorGeom {
  float x0, y0, x1, y1, z0, z1, area;
  bool  zero;
};

__device__ __forceinline__ void decode_t(int t, int& a, int& h, int& w, int& d) {
  a = t % A_;
  int s = t / A_;
  d = s % D_;
  int s2 = s / D_;
  w = s2 % W_;
  h = s2 / W_;
}

__device__ __forceinline__ AnchorGeom make_anchor(const float* __restrict__ anchors,
                                                  int a, int h, int w, int d) {
  float sx = (float)w * 16.f, sy = (float)h * 16.f, sz = (float)d * 16.f;
  const float* an = anchors + a * 6;
  AnchorGeom g;
  g.x0 = an[0] + sx; g.y0 = an[1] + sy;
  g.x1 = an[2] + sx; g.y1 = an[3] + sy;
  g.z0 = an[4] + sz; g.z1 = an[5] + sz;
  float aw = g.x1 - g.x0 + 1.f, ah = g.y1 - g.y0 + 1.f, ad = g.z1 - g.z0 + 1.f;
  g.area = aw * ah * ad;
  g.zero = (aw == 1.f) && (ah == 1.f) && (ad == 1.f);
  return g;
}

// ov_m for one (anchor, gt) pair — identical expression in both heavy kernels so
// the tie equality test reproduces bitwise.
__device__ __forceinline__ float iou_one(const AnchorGeom& a, const float* __restrict__ g,
                                         bool inside) {
  float gw = g[2] - g[0] + 1.f, gh = g[3] - g[1] + 1.f, gd = g[5] - g[4] + 1.f;
  float iw = fmaxf(fminf(a.x1, g[2]) - fmaxf(a.x0, g[0]) + 1.f, 0.f);
  float ih = fmaxf(fminf(a.y1, g[3]) - fmaxf(a.y0, g[1]) + 1.f, 0.f);
  float id = fmaxf(fminf(a.z1, g[5]) - fmaxf(a.z0, g[4]) + 1.f, 0.f);
  float inter = iw * ih * id;
  float ov = inter / (a.area + gw * gh * gd - inter);
  if ((gw == 1.f) && (gh == 1.f) && (gd == 1.f)) ov = 0.f;
  if (a.zero) ov = -1.f;
  return inside ? ov : -1.f;
}

__device__ __forceinline__ bool anchor_inside(const AnchorGeom& g,
                                              const float* __restrict__ im_info) {
  return (g.x0 >= 0.f) && (g.y0 >= 0.f) && (g.z0 >= 0.f) &&
         (g.x1 < im_info[1]) && (g.y1 < im_info[0]) && (g.z1 < im_info[2]);
}

// ---------------- kernels -----------------------------------------------------
__global__ void k_init(unsigned* __restrict__ gkey, int* __restrict__ fg_cnt,
                       int* __restrict__ bg_hist) {
  int i = blockIdx.x * blockDim.x + threadIdx.x;
  if (i < B_ * K_)   gkey[i] = 0u;
  if (i < B_)        fg_cnt[i] = 0;
  if (i < B_ * NBUCK) bg_hist[i] = 0;
}

struct ShA { float gt[GTN]; unsigned key[K_]; };   // gt MUST be first (TDM lds_addr=0)

__global__ void k_overlaps(const float* __restrict__ gt_boxes,
                           const float* __restrict__ anchors,
                           const float* __restrict__ im_info,
                           float* __restrict__ max_ov, int* __restrict__ amax,
                           unsigned* __restrict__ gkey) {
  __shared__ ShA s;
  const int b   = blockIdx.y;
  const int t   = blockIdx.x * blockDim.x + threadIdx.x;
  const int tid = threadIdx.x;

  load_gt_to_lds<GTN>(gt_boxes + (size_t)b * GTN, s.gt, tid, blockDim.x);
  if (tid < K_) s.key[tid] = 0u;
  __syncthreads();

  int a, h, w, d;
  decode_t(t, a, h, w, d);
  AnchorGeom ag = make_anchor(anchors, a, h, w, d);
  bool inside = anchor_inside(ag, im_info);

  float best = -2.f;
  int   barg = 0;
#pragma unroll
  for (int k = 0; k < K_; ++k) {
    float ovm = iou_one(ag, s.gt + k * 7, inside);
    if (ovm > best) { best = ovm; barg = k; }      // first-max == jnp.argmax
    atomicMax(&s.key[k], fkey(ovm));               // LDS reduction per gt
  }
  size_t idx = (size_t)b * TOTAL + t;
  max_ov[idx] = best;
  amax[idx]   = barg;
  __syncthreads();
  if (tid < K_) atomicMax(&gkey[b * K_ + tid], s.key[tid]);
}

struct ShB { float gt[GTN]; float gmax[K_]; };     // gt MUST be first

__global__ void k_label_targets(const float* __restrict__ gt_boxes,
                                const float* __restrict__ anchors,
                                const float* __restrict__ im_info,
                                const unsigned* __restrict__ gkey,
                                const float* __restrict__ max_ov,
                                const int* __restrict__ amax,
                                const float* __restrict__ rand_fg,
                                const float* __restrict__ rand_bg,
                                float* __restrict__ labels0,
                                float* __restrict__ labels1,
                                int* __restrict__ fg_cnt,
                                float* __restrict__ fg_r, int* __restrict__ fg_idx,
                                int* __restrict__ bg_hist,
                                float* __restrict__ out_bt) {
  __shared__ ShB s;
  const int b   = blockIdx.y;
  const int t   = blockIdx.x * blockDim.x + threadIdx.x;
  const int tid = threadIdx.x;

  load_gt_to_lds<GTN>(gt_boxes + (size_t)b * GTN, s.gt, tid, blockDim.x);
  if (tid < K_) {
    float g = funkey(gkey[b * K_ + tid]);
    s.gmax[tid] = (g == 0.f) ? 1e-5f : g;          // gt_max==0 -> 1e-5 rule
  }
  __syncthreads();

  size_t idx = (size_t)b * TOTAL + t;
  __builtin_prefetch(rand_bg + idx, 0, 0);         // global_prefetch_b8

  int a, h, w, d;
  decode_t(t, a, h, w, d);
  AnchorGeom ag = make_anchor(anchors, a, h, w, d);
  bool inside = anchor_inside(ag, im_info);

  bool tie = false;
#pragma unroll
  for (int k = 0; k < K_; ++k) {
    float ovm = iou_one(ag, s.gt + k * 7, inside);
    tie = tie || (ovm == s.gmax[k]);
  }

  float mo  = max_ov[idx];
  float lab = -1.f;
  if (inside) {
    if (mo < NEG_OV) lab = 0.f;
    if (tie || (mo >= POS_OV)) lab = 1.f;
  }
  labels0[idx] = lab;
  labels1[idx] = lab;

  if (lab == 1.f) {
    int i = atomicAdd(&fg_cnt[b], 1);
    if (i < FG_CAP) { fg_r[b * FG_CAP + i] = rand_fg[idx]; fg_idx[b * FG_CAP + i] = t; }
  } else if (lab == 0.f) {
    float r = rand_bg[idx];
    int bk = (int)(r * (float)NBUCK);
    bk = bk < NBUCK - 1 ? bk : NBUCK - 1;
    atomicAdd(&bg_hist[b * NBUCK + bk], 1);
  }

  // bbox targets (independent of sampling) -> final layout directly
  const float* g = s.gt + amax[idx] * 7;
  float ew = ag.x1 - ag.x0 + 1.f, eh = ag.y1 - ag.y0 + 1.f, ed = ag.z1 - ag.z0 + 1.f;
  float ecx = ag.x0 + 0.5f * (ew - 1.f);
  float ecy = ag.y0 + 0.5f * (eh - 1.f);
  float ecz = ag.z0 + 0.5f * (ed - 1.f);
  float gw = g[2] - g[0] + 1.f, gh = g[3] - g[1] + 1.f, gd = g[5] - g[4] + 1.f;
  float gcx = g[0] + 0.5f * (gw - 1.f);
  float gcy = g[1] + 0.5f * (gh - 1.f);
  float gcz = g[4] + 0.5f * (gd - 1.f);
  float tg[6] = { (gcx - ecx) / ew, (gcy - ecy) / eh, (gcz - ecz) / ed,
                  logf(gw / ew), logf(gh / eh), logf(gd / ed) };
#pragma unroll
  for (int j = 0; j < 6; ++j) {
    float v = inside ? tg[j] : 0.f;
    size_t oc = ((((size_t)b * (6 * A_) + (a * 6 + j)) * H_ + h) * W_ + w) * D_ + d;
    out_bt[oc] = v;
  }
}

__global__ void k_fg_sample(const int* __restrict__ fg_cnt,
                            const float* __restrict__ fg_r,
                            const int* __restrict__ fg_idx,
                            float* __restrict__ labels1, int* __restrict__ num_bg) {
  int b = blockIdx.x;
  int n_total = fg_cnt[b];
  int n = n_total < FG_CAP ? n_total : FG_CAP;
  if (n_total > NUM_FG) {
    for (int i = threadIdx.x; i < n; i += blockDim.x) {
      float ri = fg_r[b * FG_CAP + i];
      int   ti = fg_idx[b * FG_CAP + i];
      int   cnt = 0;
      for (int j = 0; j < n; ++j) {
        float rj = fg_r[b * FG_CAP + j];
        int   tj = fg_idx[b * FG_CAP + j];
        cnt += (rj > ri || (rj == ri && tj < ti)) ? 1 : 0;   // stable-sort rank
      }
      if (cnt >= NUM_FG) labels1[(size_t)b * TOTAL + ti] = -1.f;
    }
  }
  if (threadIdx.x == 0) {
    int kept = n_total > NUM_FG ? NUM_FG : n_total;
    num_bg[b] = RPN_BATCH - kept;
  }
}

__global__ void k_bg_thresh(const int* __restrict__ hist, const int* __restrict__ num_bg,
                            int* __restrict__ tb, int* __restrict__ cab) {
  int b = blockIdx.x;
  if (threadIdx.x == 0) {
    int need = num_bg[b], cum = 0, t_b = -1, ca = 0;
    for (int bk = NBUCK - 1; bk >= 0; --bk) {
      int hc = hist[b * NBUCK + bk];
      if (cum + hc >= need) { t_b = bk; ca = cum; break; }
      cum += hc;
    }
    tb[b] = t_b;   // -1 => keep all bg
    cab[b] = ca;
  }
}

__global__ void k_bg_apply(const float* __restrict__ labels0,
                           const float* __restrict__ rand_bg,
                           const int* __restrict__ tb, const int* __restrict__ cab,
                           const int* __restrict__ num_bg,
                           float* __restrict__ labels1) {
  int b = blockIdx.y;
  int t = blockIdx.x * blockDim.x + threadIdx.x;
  size_t idx = (size_t)b * TOTAL + t;
  if (labels0[idx] != 0.f) return;
  int t_b = tb[b];
  if (t_b < 0) return;
  float r = rand_bg[idx];
  int bk = (int)(r * (float)NBUCK);
  bk = bk < NBUCK - 1 ? bk : NBUCK - 1;
  if (bk > t_b) return;                              // rank < need: keep
  if (bk < t_b) { labels1[idx] = -1.f; return; }     // rank >= need: drop
  // boundary bucket: exact rank among its members
  int cnt = cab[b];
  const float* L0 = labels0 + (size_t)b * TOTAL;
  const float* RB = rand_bg + (size_t)b * TOTAL;
  for (int u = 0; u < TOTAL; ++u) {
    if (L0[u] == 0.f) {
      float ru = RB[u];
      int bu = (int)(ru * (float)NBUCK);
      bu = bu < NBUCK - 1 ? bu : NBUCK - 1;
      if (bu == t_b && (ru > r || (ru == r && u < t))) ++cnt;
    }
  }
  if (cnt >= num_bg[b]) labels1[idx] = -1.f;
}

// ---------------- num_examples reduction (WMMA rowsum, B = ones) --------------
#if __has_builtin(__builtin_amdgcn_wmma_f32_16x16x4_f32)
#define ATL_HAVE_WMMA 1
typedef float atl_v2f __attribute__((ext_vector_type(2)));
typedef float atl_v8f __attribute__((ext_vector_type(8)));
#else
#define ATL_HAVE_WMMA 0
#endif

__global__ void k_count(const float* __restrict__ labels1, float* __restrict__ num_ex) {
  __shared__ float sh[258];
  float c = 0.f;
  for (int t = threadIdx.x; t < TOTAL; t += 256)
    c += (labels1[(size_t)(B_ - 1) * TOTAL + t] >= 0.f) ? 1.f : 0.f;
  sh[threadIdx.x] = c;
  __syncthreads();
#if ATL_HAVE_WMMA
  if (threadIdx.x < 32) {                      // exactly wave 0 (wave32): EXEC all 1s
    int L  = threadIdx.x;
    int m  = L & 15;
    int kb = (L < 16) ? 0 : 2;
    int i0 = m * 4 + kb, i1 = i0 + 1;          // A[m][k] = p[m*4+k], p = 64 partials
    atl_v2f av;
    av.x = sh[i0] + sh[i0 + 64] + sh[i0 + 128] + sh[i0 + 192];
    av.y = sh[i1] + sh[i1 + 64] + sh[i1 + 128] + sh[i1 + 192];
    atl_v2f bv; bv.x = 1.f; bv.y = 1.f;        // B = ones -> D[m][n] = rowsum(A[m])
    atl_v8f cv = {0.f, 0.f, 0.f, 0.f, 0.f, 0.f, 0.f, 0.f};
    cv = __builtin_amdgcn_wmma_f32_16x16x4_f32(false, av, false, bv, (short)0, cv,
                                               false, false);
    float srow = cv[0] + cv[1] + cv[2] + cv[3] + cv[4] + cv[5] + cv[6] + cv[7];
    if (L == 0)  sh[256] = srow;               // rows 0..7  (column 0)
    if (L == 16) sh[257] = srow;               // rows 8..15 (column 0)
  }
  __syncthreads();
  if (threadIdx.x == 0) num_ex[0] = sh[256] + sh[257];
#else
  if (threadIdx.x == 0) {
    float s2 = 0.f;
    for (int i = 0; i < 256; ++i) s2 += sh[i];
    num_ex[0] = s2;
  }
#endif
}

__global__ void k_final(const float* __restrict__ labels1, const float* __restrict__ num_ex,
                        float* __restrict__ out_labels, float* __restrict__ out_iw,
                        float* __restrict__ out_ow) {
  int b = blockIdx.y;
  int t = blockIdx.x * blockDim.x + threadIdx.x;
  int a, h, w, d;
  decode_t(t, a, h, w, d);
  float lab = labels1[(size_t)b * TOTAL + t];
  size_t ol = ((((size_t)b * A_ + a) * H_ + h) * W_ + w) * D_ + d;
  out_labels[ol] = lab;
  float wgt = 1.f / num_ex[0];
  float iw  = (lab == 1.f) ? 1.f : 0.f;
  float ow  = (lab == 1.f || lab == 0.f) ? wgt : 0.f;
#pragma unroll
  for (int j = 0; j < 6; ++j) {
    size_t oc = ((((size_t)b * (6 * A_) + (a * 6 + j)) * H_ + h) * W_ + w) * D_ + d;
    out_iw[oc] = iw;
    out_ow[oc] = ow;
  }
}

// ---------------- launch ------------------------------------------------------
extern "C" void kernel_launch(void* const* d_in, const int* in_sizes, int n_in,
                              void* d_out, int out_size, void* d_ws, size_t ws_size,
                              hipStream_t stream) {
  (void)in_sizes; (void)n_in; (void)out_size; (void)ws_size;

  const float* gt_boxes = (const float*)d_in[1];
  const float* im_info  = (const float*)d_in[2];
  const float* anchors  = (const float*)d_in[4];
  const float* rand_fg  = (const float*)d_in[5];
  const float* rand_bg  = (const float*)d_in[6];

  float* out_labels = (float*)d_out;
  float* out_bt     = out_labels + (size_t)B_ * TOTAL;
  float* out_iw     = out_bt + (size_t)B_ * TOTAL * 6;
  float* out_ow     = out_iw + (size_t)B_ * TOTAL * 6;

  char* p = (char*)d_ws;
  auto take = [&](size_t bytes) -> char* {
    char* r = p;
    p += (bytes + 255) & ~(size_t)255;
    return r;
  };
  unsigned* gkey    = (unsigned*)take((size_t)B_ * K_ * 4);
  float*    max_ov  = (float*)take((size_t)B_ * TOTAL * 4);
  int*      amax    = (int*)take((size_t)B_ * TOTAL * 4);
  float*    labels0 = (float*)take((size_t)B_ * TOTAL * 4);
  float*    labels1 = (float*)take((size_t)B_ * TOTAL * 4);
  int*      fg_cnt  = (int*)take((size_t)B_ * 4);
  float*    fg_r    = (float*)take((size_t)B_ * FG_CAP * 4);
  int*      fg_idx  = (int*)take((size_t)B_ * FG_CAP * 4);
  int*      bg_hist = (int*)take((size_t)B_ * NBUCK * 4);
  int*      num_bg  = (int*)take((size_t)B_ * 4);
  int*      tbuf    = (int*)take((size_t)B_ * 4);
  int*      cab     = (int*)take((size_t)B_ * 4);
  float*    num_ex  = (float*)take(4);

  dim3 g(TOTAL / 256, B_);

  k_init<<<(B_ * NBUCK + 255) / 256, 256, 0, stream>>>(gkey, fg_cnt, bg_hist);
  k_overlaps<<<g, 256, 0, stream>>>(gt_boxes, anchors, im_info, max_ov, amax, gkey);
  k_label_targets<<<g, 256, 0, stream>>>(gt_boxes, anchors, im_info, gkey, max_ov, amax,
                                         rand_fg, rand_bg, labels0, labels1, fg_cnt,
                                         fg_r, fg_idx, bg_hist, out_bt);
  k_fg_sample<<<B_, 256, 0, stream>>>(fg_cnt, fg_r, fg_idx, labels1, num_bg);
  k_bg_thresh<<<B_, 32, 0, stream>>>(bg_hist, num_bg, tbuf, cab);
  k_bg_apply<<<g, 256, 0, stream>>>(labels0, rand_bg, tbuf, cab, num_bg, labels1);
  k_count<<<1, 256, 0, stream>>>(labels1, num_ex);
  k_final<<<g, 256, 0, stream>>>(labels1, num_ex, out_labels, out_iw, out_ow);
}